// GCNLayer_72602127172024
// MI455X (gfx1250) — compile-verified
//
#include <hip/hip_runtime.h>

// ---------------------------------------------------------------------------
// GCN layer: out[l,b,:,:] = M_l (25x25) @ x[b,l,:,:] (25x256), fp32.
// Memory-bound (6.2 FLOP/byte, ~2.1 GB traffic -> ~90us @ 23.3 TB/s), but the
// required ~146 TFLOP/s fp32 pushes past VALU: use V_WMMA_F32_16X16X4_F32.
// Pad i: 25->32 (two 16-row tiles), j(K): 25->28 (seven K=4 steps).
// One 256-thread WG (8 wave32) per (b,l); x read exactly once per WG.
// ---------------------------------------------------------------------------

typedef __attribute__((ext_vector_type(2))) float v2f;
typedef __attribute__((ext_vector_type(8))) float v8f;

#define B_ 2048
#define L_ 20
#define J_ 25
#define C_ 256
#define F_ 64
#define K_ 4

#define MI 32   // padded i (rows of M)
#define MJ 28   // padded j (K dim), 7 steps of 4

__global__ __launch_bounds__(256)
void gcn_wmma_f32_kernel(const float* __restrict__ x,
                         const float* __restrict__ A,
                         const float* __restrict__ kern,
                         float* __restrict__ out)
{
    __shared__ float wsum[K_];
    __shared__ float Mlds[MI * MJ];   // 3584 B, zero padded 32x28

    const int wg  = blockIdx.x;       // 0 .. B_*L_-1
    const int b   = wg / L_;
    const int l   = wg - b * L_;
    const int tid = threadIdx.x;

    // ---- phase 1: w[k] = sum_f kernel[f,k,l]  (deterministic fixed order) ----
    if (tid < K_) {
        float s = 0.0f;
        #pragma unroll
        for (int f = 0; f < F_; ++f)
            s += kern[f * (K_ * L_) + tid * L_ + l];
        wsum[tid] = s;
    }
    __syncthreads();

    const float w0 = wsum[0], w1 = wsum[1], w2 = wsum[2], w3 = wsum[3];

    // ---- phase 2: M[i][j] = w0 + a*(w1 + a*(w2 + a*w3)), a = A[i][j]; pad 0 ----
    for (int idx = tid; idx < MI * MJ; idx += 256) {
        const int i = idx / MJ;
        const int j = idx - i * MJ;
        float v = 0.0f;
        if (i < J_ && j < J_) {
            const float a = A[i * J_ + j];
            v = w0 + a * (w1 + a * (w2 + a * w3));
        }
        Mlds[idx] = v;
    }
    __syncthreads();

    // ---- phase 3: WMMA GEMM ----
    const int lane = tid & 31;
    const int wid  = tid >> 5;     // wave 0..7
    const int ln   = lane & 15;    // column-in-tile / row-in-tile
    const int half = lane >> 4;    // lane half selects K pair (ISA A/B layout)

    // A fragments for both i-tiles, all 7 K-steps (shared by both c-tiles).
    // ISA 32-bit A 16x4 layout: VGPR v, lane-half h -> K = kb + 2h + v  (contiguous pair)
    v2f a0[7], a1[7];
    #pragma unroll
    for (int s = 0; s < 7; ++s) {
        const int k = 4 * s + 2 * half;
        a0[s] = *(const v2f*)&Mlds[ln        * MJ + k];
        a1[s] = *(const v2f*)&Mlds[(ln + 16) * MJ + k];
    }

    const float* __restrict__ xblk = x   + (size_t)(b * L_ + l) * (J_ * C_);
    float*       __restrict__ oblk = out + (size_t)(l * B_ + b) * (J_ * C_);

    #pragma unroll
    for (int t = 0; t < 2; ++t) {
        const int cc = (wid + 8 * t) * 16 + ln;   // this lane's output column

        v8f acc0 = {};   // i-tile rows 0..15
        v8f acc1 = {};   // i-tile rows 16..31 (rows >=25 discarded)

        #pragma unroll
        for (int s = 0; s < 7; ++s) {
            const int j0 = 4 * s + 2 * half;       // B fragment: K = j0, j0+1
            v2f bf;
            bf.x = (j0     < J_) ? xblk[j0       * C_ + cc] : 0.0f;
            bf.y = (j0 + 1 < J_) ? xblk[(j0 + 1) * C_ + cc] : 0.0f;

            acc0 = __builtin_amdgcn_wmma_f32_16x16x4_f32(
                       false, a0[s], false, bf, (short)0, acc0, false, false);
            acc1 = __builtin_amdgcn_wmma_f32_16x16x4_f32(
                       false, a1[s], false, bf, (short)0, acc1, false, false);
        }

        // C/D layout: VGPR v, lane-half h -> row = v + 8h, col = ln
        #pragma unroll
        for (int v = 0; v < 8; ++v) {
            const int r0 = v + 8 * half;           // 0..15: always valid
            oblk[r0 * C_ + cc] = acc0[v];
            const int r1 = 16 + v + 8 * half;      // 16..31: keep rows < 25
            if (r1 < J_)
                oblk[r1 * C_ + cc] = acc1[v];
        }
    }
}

extern "C" void kernel_launch(void* const* d_in, const int* in_sizes, int n_in,
                              void* d_out, int out_size, void* d_ws, size_t ws_size,
                              hipStream_t stream) {
    const float* x    = (const float*)d_in[0];   // [B, L, J, C]
    const float* A    = (const float*)d_in[1];   // [J, J]
    const float* kern = (const float*)d_in[2];   // [F, K, L]
    float* out        = (float*)d_out;           // [L, B, J, C]

    (void)in_sizes; (void)n_in; (void)out_size; (void)d_ws; (void)ws_size;

    dim3 grid(B_ * L_);   // one WG per (b, l)
    dim3 block(256);
    gcn_wmma_f32_kernel<<<grid, block, 0, stream>>>(x, A, kern, out);
}